// LGNLayer_51951924413111
// MI455X (gfx1250) — compile-verified
//
#include <hip/hip_runtime.h>
#include <hip/hip_bf16.h>
#include <cstddef>

// Sizes from the reference
#define N_RET 8192
#define N_LGN 4096
#define ETA    0.1f
#define MU_WTS 2.5f

typedef float v2f __attribute__((ext_vector_type(2)));
typedef float v8f __attribute__((ext_vector_type(8)));

// ---------------------------------------------------------------------------
// Kernel 1: node_x = retina_weights @ is_firing ; new_firing = node_x > thresh
// One wave (32 lanes) computes a 16-row tile via V_WMMA_F32_16X16X4_F32,
// broadcasting the vector across all 16 N-columns (column 0 is read out).
// ---------------------------------------------------------------------------
__global__ void __launch_bounds__(256)
k_retina_matvec(const float* __restrict__ W, const float* __restrict__ x,
                const float* __restrict__ thresh, float* __restrict__ new_firing)
{
    __shared__ float xs[N_RET];              // 32 KB of the 320 KB WGP LDS
    const int t = threadIdx.x;
    for (int i = t; i < N_RET; i += 256) xs[i] = x[i];
    __syncthreads();

    const int wave = t >> 5;
    const int lane = t & 31;
    const int tile = blockIdx.x * 8 + wave;          // 512 tiles of 16 rows
    const int row16 = tile * 16;
    // ISA 32-bit A layout: lanes 0-15 -> M=lane, K-pair {0,1}; lanes 16-31 -> M=lane-16, K-pair {2,3}
    const int mrow = row16 + (lane & 15);
    const int koff = (lane >> 4) * 2;
    const float* __restrict__ arow = W + (size_t)mrow * N_RET + koff;

    v8f acc = {};
    for (int k = 0; k < N_RET; k += 32) {
        __builtin_prefetch(arow + k + 1024, 0, 3);   // global_prefetch_b8, per-lane row stream
#pragma unroll
        for (int j = 0; j < 8; ++j) {
            const int kk = k + 4 * j;
            v2f a = *(const v2f*)(arow + kk);        // global_load_b64
            v2f b = *(const v2f*)(&xs[kk + koff]);   // ds_load_b64 (broadcast across N)
            acc = __builtin_amdgcn_wmma_f32_16x16x4_f32(
                      false, a, false, b, (short)0, acc, false, false);
        }
    }

    // Column N=0 of the 16x16 f32 D tile: lane 0 holds M=0..7, lane 16 holds M=8..15
    if (lane == 0 || lane == 16) {
        const int base = row16 + ((lane >> 4) << 3);
#pragma unroll
        for (int m = 0; m < 8; ++m) {
            const float nx = acc[m];
            new_firing[base + m] = (nx > thresh[base + m]) ? 1.0f : 0.0f;
        }
    }
}

// ---------------------------------------------------------------------------
// Kernel 2: lgn_act = relu(lgn_weights @ new_firing)
// ---------------------------------------------------------------------------
__global__ void __launch_bounds__(256)
k_lgn_matvec(const float* __restrict__ W, const float* __restrict__ f,
             float* __restrict__ lgn_act)
{
    __shared__ float xs[N_RET];
    const int t = threadIdx.x;
    for (int i = t; i < N_RET; i += 256) xs[i] = f[i];
    __syncthreads();

    const int wave = t >> 5;
    const int lane = t & 31;
    const int tile = blockIdx.x * 8 + wave;          // 256 tiles of 16 rows
    const int row16 = tile * 16;
    const int mrow = row16 + (lane & 15);
    const int koff = (lane >> 4) * 2;
    const float* __restrict__ arow = W + (size_t)mrow * N_RET + koff;

    v8f acc = {};
    for (int k = 0; k < N_RET; k += 32) {
        __builtin_prefetch(arow + k + 1024, 0, 3);
#pragma unroll
        for (int j = 0; j < 8; ++j) {
            const int kk = k + 4 * j;
            v2f a = *(const v2f*)(arow + kk);
            v2f b = *(const v2f*)(&xs[kk + koff]);
            acc = __builtin_amdgcn_wmma_f32_16x16x4_f32(
                      false, a, false, b, (short)0, acc, false, false);
        }
    }

    if (lane == 0 || lane == 16) {
        const int base = row16 + ((lane >> 4) << 3);
#pragma unroll
        for (int m = 0; m < 8; ++m)
            lgn_act[base + m] = fmaxf(acc[m], 0.0f);
    }
}

// ---------------------------------------------------------------------------
// Kernel 3: act = relu(lgn_act - thr); (val, idx) = first-occurrence argmax
// Single block, 1024 threads, LDS tree reduction.
// ---------------------------------------------------------------------------
__global__ void __launch_bounds__(1024)
k_argmax(const float* __restrict__ lgn_act, const float* __restrict__ thr,
         float* __restrict__ wsF, int* __restrict__ wsI)
{
    __shared__ float sv[1024];
    __shared__ int   si[1024];
    const int t = threadIdx.x;
    float best = -1.0f;
    int   bidx = 0x7fffffff;
    for (int i = t; i < N_LGN; i += 1024) {
        float a = fmaxf(lgn_act[i] - thr[i], 0.0f);
        if (a > best || (a == best && i < bidx)) { best = a; bidx = i; }
    }
    sv[t] = best; si[t] = bidx;
    __syncthreads();
    for (int s = 512; s > 0; s >>= 1) {
        if (t < s) {
            float ov = sv[t + s]; int oi = si[t + s];
            if (ov > sv[t] || (ov == sv[t] && oi < si[t])) { sv[t] = ov; si[t] = oi; }
        }
        __syncthreads();
    }
    if (t == 0) { wsF[0] = sv[0]; wsI[1] = si[0]; }
}

// ---------------------------------------------------------------------------
// Kernel 4a: sum the updated winner row -> scale = MU*N/sum ; update thresholds
// ---------------------------------------------------------------------------
__global__ void __launch_bounds__(1024)
k_rownorm(const float* __restrict__ lgnW, const float* __restrict__ firing,
          const float* __restrict__ thr_in, float* __restrict__ thr_out,
          float* __restrict__ wsF, const int* __restrict__ wsI)
{
    __shared__ float ssum[1024];
    const int t = threadIdx.x;
    const float val = wsF[0];
    const int   idx = wsI[1];
    const bool fire = val > 0.0f;

    for (int i = t; i < N_LGN; i += 1024)
        thr_out[i] = thr_in[i] + ((fire && i == idx) ? 0.005f * val : 0.0f);

    const float* __restrict__ row = lgnW + (size_t)idx * N_RET;
    float s = 0.0f;
    for (int i = t; i < N_RET; i += 1024)
        s += row[i] + ETA * val * firing[i];
    ssum[t] = s;
    __syncthreads();
    for (int st = 512; st > 0; st >>= 1) {
        if (t < st) ssum[t] += ssum[t + st];
        __syncthreads();
    }
    if (t == 0)
        wsF[2] = fire ? (MU_WTS * (float)N_RET / ssum[0]) : 1.0f;
}

// ---------------------------------------------------------------------------
// Kernel 4b: 128 MB weight copy (float4, grid-strided) patching the winner row
// ---------------------------------------------------------------------------
__global__ void __launch_bounds__(256)
k_copy_update(const float* __restrict__ lgnW, const float* __restrict__ firing,
              float* __restrict__ outW, const float* __restrict__ wsF,
              const int* __restrict__ wsI)
{
    const float val   = wsF[0];
    const int   idx   = wsI[1];
    const float scale = wsF[2];
    const bool  fire  = val > 0.0f;

    const size_t n4 = (size_t)N_LGN * N_RET / 4;            // 8M float4
    const size_t stride = (size_t)gridDim.x * blockDim.x;
    for (size_t i = (size_t)blockIdx.x * blockDim.x + threadIdx.x; i < n4; i += stride) {
        float4 w = ((const float4*)lgnW)[i];
        const int row = (int)(i >> 11);                     // 2048 float4 per row
        if (fire && row == idx) {
            float4 fv = ((const float4*)firing)[i & 2047];
            w.x = (w.x + ETA * val * fv.x) * scale;
            w.y = (w.y + ETA * val * fv.y) * scale;
            w.z = (w.z + ETA * val * fv.z) * scale;
            w.w = (w.w + ETA * val * fv.w) * scale;
        }
        ((float4*)outW)[i] = w;
    }
}

// ---------------------------------------------------------------------------
// Host-side launcher (graph-capture safe: only async launches on `stream`)
// ---------------------------------------------------------------------------
extern "C" void kernel_launch(void* const* d_in, const int* in_sizes, int n_in,
                              void* d_out, int out_size, void* d_ws, size_t ws_size,
                              hipStream_t stream)
{
    const float* is_firing     = (const float*)d_in[0];   // (8192,)
    const float* retina_w      = (const float*)d_in[1];   // (8192,8192)
    const float* node_thresh   = (const float*)d_in[2];   // (8192,)
    const float* lgn_w         = (const float*)d_in[3];   // (4096,8192)
    const float* lgn_thresh    = (const float*)d_in[4];   // (4096,)

    float* o          = (float*)d_out;
    float* new_firing = o;                                  // 8192
    float* lgn_act    = o + N_RET;                          // 4096
    float* new_w      = o + N_RET + N_LGN;                  // 4096*8192
    float* new_thr    = o + N_RET + N_LGN + (size_t)N_LGN * N_RET; // 4096

    float* wsF = (float*)d_ws;
    int*   wsI = (int*)d_ws;

    // 1) retina mat-vec + fire threshold (512 wave-tiles -> 64 blocks x 8 waves)
    k_retina_matvec<<<64, 256, 0, stream>>>(retina_w, is_firing, node_thresh, new_firing);
    // 2) LGN mat-vec + relu (256 wave-tiles -> 32 blocks x 8 waves)
    k_lgn_matvec<<<32, 256, 0, stream>>>(lgn_w, new_firing, lgn_act);
    // 3) winner-take-all argmax
    k_argmax<<<1, 1024, 0, stream>>>(lgn_act, lgn_thresh, wsF, wsI);
    // 4a) winner-row normalization factor + threshold output
    k_rownorm<<<1, 1024, 0, stream>>>(lgn_w, new_firing, lgn_thresh, new_thr, wsF, wsI);
    // 4b) bulk weight copy with in-flight winner-row Hebbian update
    k_copy_update<<<2048, 256, 0, stream>>>(lgn_w, new_firing, new_w, wsF, wsI);
}